// ConfusionWeightedBhatReg_68985764708474
// MI455X (gfx1250) — compile-verified
//
#include <hip/hip_runtime.h>
#include <math.h>

// ConfusionWeightedBhatReg for MI455X (gfx1250, wave32).
// One-hot scatter reductions are cast as Hᵀ·Z GEMMs on the exact-precision
// V_WMMA_F32_16X16X4_F32 path; pairwise Bhattacharyya runs on VALU/trans.

typedef __attribute__((ext_vector_type(2))) float v2f;
typedef __attribute__((ext_vector_type(8))) float v8f;

#define BB 16384
#define DD 384
#define CC 512
#define TOPM 64
#define NPAIRS ((CC * (CC - 1)) / 2)   // 130816
#define EPSF 1e-6f

// ---------------------------------------------------------------- utilities
__global__ void zero_counts_kernel(float* counts) {
    int i = blockIdx.x * blockDim.x + threadIdx.x;
    if (i < CC) counts[i] = 0.0f;
}

__global__ void count_kernel(const int* __restrict__ y, float* counts) {
    int i = blockIdx.x * blockDim.x + threadIdx.x;
    if (i < BB) atomicAdd(&counts[y[i]], 1.0f);
}

// ---------------------------------------------------------------- softmax
// One 128-thread block per row of logits (512 cols).
__global__ void softmax_kernel(const float* __restrict__ logits, float* __restrict__ P) {
    const int row = blockIdx.x;
    const int tid = threadIdx.x;
    const float* L = logits + (size_t)row * CC;
    float* O = P + (size_t)row * CC;
    __shared__ float red[128];

    float m = -3.402823466e38f;
    for (int c = tid; c < CC; c += 128) m = fmaxf(m, L[c]);
    red[tid] = m; __syncthreads();
    for (int s = 64; s > 0; s >>= 1) { if (tid < s) red[tid] = fmaxf(red[tid], red[tid + s]); __syncthreads(); }
    m = red[0]; __syncthreads();

    float sum = 0.0f;
    for (int c = tid; c < CC; c += 128) { float e = expf(L[c] - m); O[c] = e; sum += e; }
    red[tid] = sum; __syncthreads();
    for (int s = 64; s > 0; s >>= 1) { if (tid < s) red[tid] += red[tid + s]; __syncthreads(); }
    const float inv = 1.0f / red[0];
    for (int c = tid; c < CC; c += 128) O[c] *= inv;
}

// ---------------------------------------------------------------- WMMA GEMMs
// sum_z[c,d] = sum_b 1[y[b]==c] * z[b,d]; sum_z2 with z^2.  A = one-hot tile
// (exact in f32), B = z tile, accumulate on V_WMMA_F32_16X16X4_F32.
// One wave per (16-class, 16-dim) output tile, K swept 4 samples at a time.
__global__ void stats_wmma_kernel(const float* __restrict__ z0,
                                  const float* __restrict__ z1,
                                  const float* __restrict__ z2,
                                  const int* __restrict__ y,
                                  float* __restrict__ sumz,
                                  float* __restrict__ sumz2) {
    const int ct = blockIdx.x;            // class tile  (32)
    const int dt = blockIdx.y;            // dim tile    (24)
    const int layer = blockIdx.z;         // layer       (3)
    const float* Z = (layer == 0) ? z0 : (layer == 1) ? z1 : z2;

    const int lane = threadIdx.x;         // wave32
    const int half = lane >> 4;
    const int lm   = lane & 15;
    const int c0 = ct * 16, d0 = dt * 16;
    const int kb = half * 2;              // ISA A/B f32 lane layout: K={0,1} / {2,3}

    v8f acc1 = {};
    v8f acc2 = {};
    for (int b0 = 0; b0 < BB; b0 += 4) {
        const int ya = y[b0 + kb];
        const int yb = y[b0 + kb + 1];
        v2f a;                             // one-hot A fragment (16x4), exact
        a.x = (ya == c0 + lm) ? 1.0f : 0.0f;
        a.y = (yb == c0 + lm) ? 1.0f : 0.0f;
        const float zx = Z[(size_t)(b0 + kb) * DD + d0 + lm];
        const float zy = Z[(size_t)(b0 + kb + 1) * DD + d0 + lm];
        v2f bz;  bz.x  = zx;      bz.y  = zy;
        v2f bz2; bz2.x = zx * zx; bz2.y = zy * zy;
        acc1 = __builtin_amdgcn_wmma_f32_16x16x4_f32(false, a, false, bz,  (short)0, acc1, false, false);
        acc2 = __builtin_amdgcn_wmma_f32_16x16x4_f32(false, a, false, bz2, (short)0, acc2, false, false);
    }
    float* S1 = sumz  + (size_t)layer * CC * DD;
    float* S2 = sumz2 + (size_t)layer * CC * DD;
#pragma unroll
    for (int v = 0; v < 8; ++v) {
        const int m = v + 8 * half;       // C/D tile layout (ISA 7.12.2)
        S1[(size_t)(c0 + m) * DD + d0 + lm] = acc1[v];
        S2[(size_t)(c0 + m) * DD + d0 + lm] = acc2[v];
    }
}

// sum_p[c,n] = sum_b 1[y[b]==c] * P[b,n]   (same structure, N over 512 cols)
__global__ void sump_wmma_kernel(const float* __restrict__ P,
                                 const int* __restrict__ y,
                                 float* __restrict__ sump) {
    const int ct = blockIdx.x, nt = blockIdx.y;
    const int lane = threadIdx.x;
    const int half = lane >> 4, lm = lane & 15;
    const int c0 = ct * 16, n0 = nt * 16;
    const int kb = half * 2;

    v8f acc = {};
    for (int b0 = 0; b0 < BB; b0 += 4) {
        const int ya = y[b0 + kb];
        const int yb = y[b0 + kb + 1];
        v2f a;
        a.x = (ya == c0 + lm) ? 1.0f : 0.0f;
        a.y = (yb == c0 + lm) ? 1.0f : 0.0f;
        v2f bp;
        bp.x = P[(size_t)(b0 + kb) * CC + n0 + lm];
        bp.y = P[(size_t)(b0 + kb + 1) * CC + n0 + lm];
        acc = __builtin_amdgcn_wmma_f32_16x16x4_f32(false, a, false, bp, (short)0, acc, false, false);
    }
#pragma unroll
    for (int v = 0; v < 8; ++v) {
        const int m = v + 8 * half;
        sump[(size_t)(c0 + m) * CC + n0 + lm] = acc[v];
    }
}

// ---------------------------------------------------------------- alpha
__global__ void alpha_kernel(const float* __restrict__ sump,
                             const float* __restrict__ counts,
                             float* __restrict__ alpha) {
    int idx = blockIdx.x * blockDim.x + threadIdx.x;
    if (idx >= CC * CC) return;
    int i = idx / CC, j = idx % CC;
    float mij = sump[(size_t)i * CC + j] / fmaxf(counts[i], 1.0f);
    float mji = sump[(size_t)j * CC + i] / fmaxf(counts[j], 1.0f);
    alpha[idx] = (i == j) ? 0.0f : 0.5f * (mij + mji);
}

// ---------------------------------------------------------------- mu/var/slv
// In-place: sumz -> mu, sumz2 -> var. slv[l,c] = sum_d log(var+eps).
__global__ void finalize_stats_kernel(const float* __restrict__ counts,
                                      float* __restrict__ sumz,
                                      float* __restrict__ sumz2,
                                      float* __restrict__ slv) {
    const int c = blockIdx.x, layer = blockIdx.y, tid = threadIdx.x;
    const float dn = fmaxf(counts[c], 1.0f);
    float* MU = sumz  + ((size_t)layer * CC + c) * DD;
    float* VA = sumz2 + ((size_t)layer * CC + c) * DD;
    float ls = 0.0f;
    for (int d = tid; d < DD; d += 128) {
        float mu = MU[d] / dn;
        float va = fmaxf(VA[d] / dn - mu * mu, EPSF);
        MU[d] = mu; VA[d] = va;
        ls += logf(va + EPSF);
    }
    __shared__ float red[128];
    red[tid] = ls; __syncthreads();
    for (int s = 64; s > 0; s >>= 1) { if (tid < s) red[tid] += red[tid + s]; __syncthreads(); }
    if (tid == 0) slv[layer * CC + c] = red[0];
}

// ---------------------------------------------------------------- Bhattacharyya
// One wave per (i,j) pair (j>i), 384 dims across 32 lanes, shfl_xor reduce.
__global__ void rho_kernel(const float* __restrict__ mu_all,
                           const float* __restrict__ var_all,
                           const float* __restrict__ slv,
                           const float* __restrict__ counts,
                           const float* __restrict__ alpha,
                           float* __restrict__ aArr,
                           float* __restrict__ rArr) {
    const int i = blockIdx.x, j = blockIdx.y, layer = blockIdx.z;
    if (j <= i) return;
    const int lane = threadIdx.x;
    const float* mi = mu_all  + ((size_t)layer * CC + i) * DD;
    const float* mj = mu_all  + ((size_t)layer * CC + j) * DD;
    const float* vi = var_all + ((size_t)layer * CC + i) * DD;
    const float* vj = var_all + ((size_t)layer * CC + j) * DD;
    float t1 = 0.0f, tl = 0.0f;
    for (int d = lane; d < DD; d += 32) {
        float va = 0.5f * (vi[d] + vj[d]) + EPSF;
        float df = mi[d] - mj[d];
        t1 += df * df / va;
        tl += logf(va);
    }
    for (int off = 16; off > 0; off >>= 1) {
        t1 += __shfl_xor(t1, off, 32);
        tl += __shfl_xor(tl, off, 32);
    }
    if (lane == 0) {
        float dist = fmaxf(0.125f * t1 +
                           0.5f * (tl - 0.5f * (slv[layer * CC + i] + slv[layer * CC + j])), 0.0f);
        float rho = expf(-dist);
        bool valid = (counts[i] >= 2.0f) && (counts[j] >= 2.0f);
        int t = i * (CC - 1) - (i * (i - 1)) / 2 + (j - i - 1);   // np.triu order
        rArr[(size_t)layer * NPAIRS + t] = valid ? rho : 0.0f;
        if (layer == 0) aArr[t] = valid ? alpha[(size_t)i * CC + j] : 0.0f;
    }
}

// ---------------------------------------------------------------- top-64
// alpha/kept are layer-independent -> one selection shared by all 3 layers.
// Iterative argmax, tie -> lowest index (matches jax.lax.top_k).
__global__ void topk_kernel(float* __restrict__ aArr,
                            float* __restrict__ a_top,
                            int* __restrict__ idx_top) {
    __shared__ float sv[256];
    __shared__ int   si[256];
    const int tid = threadIdx.x;
    for (int k = 0; k < TOPM; ++k) {
        float bv = -1.0f; int bi = 0x7fffffff;
        for (int t = tid; t < NPAIRS; t += 256) {
            float v = aArr[t];
            if (v > bv || (v == bv && t < bi)) { bv = v; bi = t; }
        }
        sv[tid] = bv; si[tid] = bi; __syncthreads();
        for (int s = 128; s > 0; s >>= 1) {
            if (tid < s) {
                if (sv[tid + s] > sv[tid] || (sv[tid + s] == sv[tid] && si[tid + s] < si[tid])) {
                    sv[tid] = sv[tid + s]; si[tid] = si[tid + s];
                }
            }
            __syncthreads();
        }
        if (tid == 0) { a_top[k] = sv[0]; idx_top[k] = si[0]; aArr[si[0]] = -1.0f; }
        __syncthreads();
    }
}

// ---------------------------------------------------------------- combine
__global__ void loss_kernel(const float* __restrict__ counts,
                            const float* __restrict__ a_top,
                            const int* __restrict__ idx_top,
                            const float* __restrict__ rArr,
                            float* __restrict__ out) {
    __shared__ float red[256];
    const int tid = threadIdx.x;
    float k = 0.0f;
    for (int c = tid; c < CC; c += 256) k += (counts[c] >= 2.0f) ? 1.0f : 0.0f;
    red[tid] = k; __syncthreads();
    for (int s = 128; s > 0; s >>= 1) { if (tid < s) red[tid] += red[tid + s]; __syncthreads(); }
    if (tid == 0) {
        const float keptsum = red[0];
        float sa = 0.0f;
        for (int q = 0; q < TOPM; ++q) sa += a_top[q];
        float total = 0.0f;
        for (int l = 0; l < 3; ++l) {
            float sl = 0.0f;
            for (int q = 0; q < TOPM; ++q)
                sl += a_top[q] * rArr[(size_t)l * NPAIRS + idx_top[q]];
            total += sl / fmaxf(sa, EPSF);
        }
        out[0] = (keptsum >= 2.0f) ? (total / 3.0f) : 0.0f;
    }
}

// ---------------------------------------------------------------- launch
extern "C" void kernel_launch(void* const* d_in, const int* in_sizes, int n_in,
                              void* d_out, int out_size, void* d_ws, size_t ws_size,
                              hipStream_t stream) {
    (void)in_sizes; (void)n_in; (void)out_size; (void)ws_size;
    const float* f0     = (const float*)d_in[0];
    const float* f1     = (const float*)d_in[1];
    const float* f2     = (const float*)d_in[2];
    const float* logits = (const float*)d_in[3];
    const int*   y      = (const int*)d_in[4];
    float* out = (float*)d_out;

    float* ws = (float*)d_ws;
    size_t off = 0;
    float* P      = ws + off; off += (size_t)BB * CC;       // softmax, 33.5 MB
    float* counts = ws + off; off += CC;
    float* sumz   = ws + off; off += 3ull * CC * DD;        // -> mu
    float* sumz2  = ws + off; off += 3ull * CC * DD;        // -> var
    float* sump   = ws + off; off += (size_t)CC * CC;
    float* alpha  = ws + off; off += (size_t)CC * CC;
    float* slv    = ws + off; off += 3 * CC;
    float* aArr   = ws + off; off += NPAIRS;
    float* rArr   = ws + off; off += 3ull * NPAIRS;
    float* a_top  = ws + off; off += TOPM;
    int*   idx_top = (int*)(ws + off); off += TOPM;

    zero_counts_kernel<<<2, 256, 0, stream>>>(counts);
    count_kernel<<<BB / 256, 256, 0, stream>>>(y, counts);
    softmax_kernel<<<BB, 128, 0, stream>>>(logits, P);
    stats_wmma_kernel<<<dim3(CC / 16, DD / 16, 3), 32, 0, stream>>>(f0, f1, f2, y, sumz, sumz2);
    sump_wmma_kernel<<<dim3(CC / 16, CC / 16), 32, 0, stream>>>(P, y, sump);
    alpha_kernel<<<(CC * CC + 255) / 256, 256, 0, stream>>>(sump, counts, alpha);
    finalize_stats_kernel<<<dim3(CC, 3), 128, 0, stream>>>(counts, sumz, sumz2, slv);
    rho_kernel<<<dim3(CC, CC, 3), 32, 0, stream>>>(sumz, sumz2, slv, counts, alpha, aArr, rArr);
    topk_kernel<<<1, 256, 0, stream>>>(aArr, a_top, idx_top);
    loss_kernel<<<1, 256, 0, stream>>>(counts, a_top, idx_top, rArr, out);
}